// LocalGlobalRegistration_9483287789589
// MI455X (gfx1250) — compile-verified
//
#include <hip/hip_runtime.h>
#include <hip/hip_bf16.h>
#include <stdint.h>

#define RS        64
#define TILE      (RS * RS)          // 4096 floats per batch
#define NBINS     2048
#define KBASE     0x3F000u           // key of 0.5f  (bits >> 12)
#define KMAX      (KBASE + NBINS - 1)
#define NUM_CORR  2000
#define CAP       16384              // candidate capacity (expected ~4096)

// ---- workspace layout (u32 indices) ----
// [0 .. 2047]  global histogram
// [2048]       candidate counter
// [2049]       threshold bin key
// [2050]       threshold value T (float bits)
// [2112 .. 2112+CAP)  candidate values (float)
#define WS_HEADER 2112

typedef __attribute__((ext_vector_type(2))) float v2f;
typedef __attribute__((ext_vector_type(8))) float v8f;

// ---------------- CDNA5 async global -> LDS copy (ASYNCcnt path) ----------------
__device__ __forceinline__ void async_g2lds_b128(unsigned lds_off, const void* gptr) {
  unsigned long long ga = (unsigned long long)(uintptr_t)gptr;
  asm volatile("global_load_async_to_lds_b128 %0, %1, off"
               :: "v"(lds_off), "v"(ga)
               : "memory");
}
__device__ __forceinline__ void wait_async0() {
#if __has_builtin(__builtin_amdgcn_s_wait_asynccnt)
  __builtin_amdgcn_s_wait_asynccnt(0);
#else
  asm volatile("s_wait_asynccnt 0x0" ::: "memory");
#endif
}

// sortable bin key for positive floats; 0 if out of tracked range (< 0.5)
__device__ __forceinline__ unsigned bin_key(float v) {
  unsigned b = __float_as_uint(v);
  if ((int)b < (int)0x3F000000) return 0u;   // v < 0.5 or negative
  unsigned k = b >> 12;
  return k > KMAX ? KMAX : k;
}

__device__ __forceinline__ void top3_update(float v, float& m1, float& m2, float& m3) {
  if (v > m1)      { m3 = m2; m2 = m1; m1 = v; }
  else if (v > m2) { m3 = m2; m2 = v; }
  else if (v > m3) { m3 = v; }
}

// -------- K0: zero workspace header --------
__global__ void k_zero(unsigned* ws) {
  int i = blockIdx.x * blockDim.x + threadIdx.x;
  if (i < WS_HEADER) ws[i] = 0u;
}

// -------- K1: global histogram of values >= 0.5 (LDS-first) --------
__global__ void k_hist(const float* __restrict__ score, unsigned* __restrict__ ws) {
  __shared__ unsigned lh[NBINS];
  const int tid = threadIdx.x;
  for (int i = tid; i < NBINS; i += 256) lh[i] = 0u;
  __syncthreads();

  const float4* p = (const float4*)score;
  size_t gtid   = (size_t)blockIdx.x * 256 + tid;
  size_t stride = (size_t)gridDim.x * 256;
  for (int j = 0; j < 4; ++j) {
    float4 v = p[gtid + (size_t)j * stride];
    unsigned k;
    k = bin_key(v.x); if (k >= KBASE) atomicAdd(&lh[k - KBASE], 1u);
    k = bin_key(v.y); if (k >= KBASE) atomicAdd(&lh[k - KBASE], 1u);
    k = bin_key(v.z); if (k >= KBASE) atomicAdd(&lh[k - KBASE], 1u);
    k = bin_key(v.w); if (k >= KBASE) atomicAdd(&lh[k - KBASE], 1u);
  }
  __syncthreads();
  for (int i = tid; i < NBINS; i += 256) {
    unsigned c = lh[i];
    if (c) atomicAdd(&ws[i], c);
  }
}

// -------- K2: suffix scan -> threshold bin key --------
__global__ void k_scan(unsigned* ws) {
  if (threadIdx.x == 0 && blockIdx.x == 0) {
    unsigned total = 0, tb = 0;
    for (int k = NBINS - 1; k >= 0; --k) {
      total += ws[k];
      if (total >= (unsigned)NUM_CORR) { tb = (unsigned)k; break; }
    }
    ws[2049] = KBASE + tb;
  }
}

// -------- K3: gather candidates with key >= threshold bin --------
__global__ void k_gather(const float* __restrict__ score, unsigned* __restrict__ ws) {
  const unsigned tbKey = ws[2049];
  float* cand = (float*)(ws + WS_HEADER);
  const float4* p = (const float4*)score;
  size_t gtid   = (size_t)blockIdx.x * 256 + threadIdx.x;
  size_t stride = (size_t)gridDim.x * 256;
  for (int j = 0; j < 4; ++j) {
    float4 v = p[gtid + (size_t)j * stride];
    float a[4] = { v.x, v.y, v.z, v.w };
#pragma unroll
    for (int t = 0; t < 4; ++t) {
      if (bin_key(a[t]) >= tbKey) {
        unsigned pos = atomicAdd(&ws[2048], 1u);
        if (pos < CAP) cand[pos] = a[t];
      }
    }
  }
}

// -------- K4: single-block bitonic sort (desc) in LDS -> exact 2000th value --------
__global__ void k_select(unsigned* __restrict__ ws) {
  __shared__ float sh[CAP];
  const int tid = threadIdx.x;
  unsigned n = ws[2048];
  if (n > CAP) n = CAP;
  const float* cand = (const float*)(ws + WS_HEADER);
  for (int i = tid; i < CAP; i += 1024) sh[i] = (i < (int)n) ? cand[i] : -1.0f;
  __syncthreads();

  for (int k = 2; k <= CAP; k <<= 1) {
    for (int j = k >> 1; j > 0; j >>= 1) {
      for (int i = tid; i < CAP; i += 1024) {
        int ixj = i ^ j;
        if (ixj > i) {
          float a = sh[i], b = sh[ixj];
          bool desc = ((i & k) == 0);
          if (desc ? (a < b) : (a > b)) { sh[i] = b; sh[ixj] = a; }
        }
      }
      __syncthreads();
    }
  }
  if (tid == 0) ws[2050] = __float_as_uint(sh[NUM_CORR - 1]);
}

// -------- K5: fused output pass --------
// async tile -> LDS; WMMA outer-product builds mask_mat tile (ref ⊗ src);
// top-3 per row/col in LDS; write corr_mat and out_float gated by the mask tile.
__global__ void k_out(const float* __restrict__ score,
                      const unsigned char* __restrict__ refm,
                      const unsigned char* __restrict__ srcm,
                      const unsigned* __restrict__ ws,
                      float* __restrict__ out0, float* __restrict__ out1) {
  __shared__ __align__(16) float tile[TILE];
  __shared__ float maskT[TILE];
  __shared__ float refF[RS];
  __shared__ float srcF[RS];
  __shared__ float rowT[RS];
  __shared__ float colT[RS];
  const int b = blockIdx.x, tid = threadIdx.x;
  const float* base = score + (size_t)b * TILE;

  // stage 16KB tile via CDNA5 async global->LDS DMA (4 x b128 per thread)
#pragma unroll
  for (int j = 0; j < 4; ++j) {
    int off = (tid + j * 256) * 4;   // float index, 16B aligned
    async_g2lds_b128((unsigned)(uintptr_t)(tile + off), base + off);
  }
  // overlap: fetch the two mask vectors while the DMA runs
  if (tid < RS)            refF[tid]       = refm[(size_t)b * RS + tid] ? 1.0f : 0.0f;
  else if (tid < 2 * RS)   srcF[tid - RS]  = srcm[(size_t)b * RS + (tid - RS)] ? 1.0f : 0.0f;

  wait_async0();
  __syncthreads();

  if (tid < RS) {                       // per-row 3rd largest (rotated to avoid bank conflicts)
    float m1 = -2.f, m2 = -2.f, m3 = -2.f;
    for (int s = 0; s < RS; ++s) {
      float v = tile[tid * RS + ((s + tid) & (RS - 1))];
      top3_update(v, m1, m2, m3);
    }
    rowT[tid] = m3;
  } else if (tid < 2 * RS) {            // per-col 3rd largest (consecutive lanes -> distinct banks)
    int c = tid - RS;
    float m1 = -2.f, m2 = -2.f, m3 = -2.f;
    for (int r = 0; r < RS; ++r) top3_update(tile[r * RS + c], m1, m2, m3);
    colT[c] = m3;
  }

  // mask_mat tile = refF ⊗ srcF via V_WMMA_F32_16X16X4_F32 (rank-1, K=0 slice only).
  // 16 subtiles of 16x16; 8 waves -> 2 subtiles per wave. EXEC is full here.
  {
    const int wid = tid >> 5, lane = tid & 31;
#pragma unroll
    for (int t = 0; t < 2; ++t) {
      const int st = wid * 2 + t;           // 0..15
      const int i = st >> 2, j = st & 3;    // subtile row/col
#if __has_builtin(__builtin_amdgcn_wmma_f32_16x16x4_f32)
      // A(16x4): lanes 0-15 hold M=lane, VGPR0=K0,VGPR1=K1; lanes 16-31 hold K2,K3 -> zero
      v2f a; a.x = (lane < 16) ? refF[i * 16 + lane] : 0.0f; a.y = 0.0f;
      // B(4x16): mirrored layout; only K=0 row non-zero
      v2f bb; bb.x = (lane < 16) ? srcF[j * 16 + lane] : 0.0f; bb.y = 0.0f;
      v8f c = {};
      v8f d = __builtin_amdgcn_wmma_f32_16x16x4_f32(false, a, false, bb,
                                                    (short)0, c, false, false);
      // C/D layout: VGPR v, lane L -> (M = v + 8*(L>=16), N = L&15)
      const int Mb = i * 16 + ((lane >= 16) ? 8 : 0);
      const int N  = j * 16 + (lane & 15);
#pragma unroll
      for (int v = 0; v < 8; ++v) maskT[(Mb + v) * RS + N] = d[v];
#else
      // scalar fallback: each lane fills 8 entries of the subtile
      const int Mb = i * 16 + ((lane >= 16) ? 8 : 0);
      const int N  = j * 16 + (lane & 15);
      for (int v = 0; v < 8; ++v)
        maskT[(Mb + v) * RS + N] = refF[Mb + v - i * 16 + i * 16] * srcF[N];
#endif
    }
  }
  __syncthreads();

  const float T = __uint_as_float(ws[2050]);
  const size_t ob = (size_t)b * TILE;
#pragma unroll
  for (int j = 0; j < 16; ++j) {
    int e = tid + j * 256;
    float v = tile[e];
    int r = e >> 6, s = e & (RS - 1);
    float m = maskT[e];                             // mask_mat (0/1)
    bool g = (v >= T);                              // global top-2000
    float o = (v >= rowT[r] ? v : 0.f)              // ref_score_mat
            + (v >= colT[s] ? v : 0.f)              // src_score_mat
            + (g ? v : 0.f);                        // sel_score_mat
    out0[ob + e] = (g ? 1.0f : 0.0f) * m;           // corr_mat & mask_mat
    out1[ob + e] = o * m;                           // out_float * mask
  }
}

extern "C" void kernel_launch(void* const* d_in, const int* in_sizes, int n_in,
                              void* d_out, int out_size, void* d_ws, size_t ws_size,
                              hipStream_t stream) {
  const float* score = (const float*)d_in[0];
  const unsigned char* refm = (const unsigned char*)d_in[1];
  const unsigned char* srcm = (const unsigned char*)d_in[2];
  unsigned* ws = (unsigned*)d_ws;
  float* out = (float*)d_out;
  const int nb = in_sizes[0] / TILE;        // 4096 batches
  const size_t half = (size_t)out_size / 2; // corr_mat ++ out_float

  k_zero  <<<(WS_HEADER + 255) / 256, 256, 0, stream>>>(ws);
  k_hist  <<<nb, 256, 0, stream>>>(score, ws);
  k_scan  <<<1, 32, 0, stream>>>(ws);
  k_gather<<<nb, 256, 0, stream>>>(score, ws);
  k_select<<<1, 1024, 0, stream>>>(ws);
  k_out   <<<nb, 256, 0, stream>>>(score, refm, srcm, ws, out, out + half);
}